// BaseGNNRecommender_85727547228235
// MI455X (gfx1250) — compile-verified
//
#include <hip/hip_runtime.h>
#include <hip/hip_bf16.h>
#include <math.h>

typedef __attribute__((ext_vector_type(2))) float v2f;
typedef __attribute__((ext_vector_type(8))) float v8f;

#define NUM_USERS   200000
#define NUM_PRODS   100000
#define N_NODES     (NUM_USERS + NUM_PRODS)
#define EDIM        64

static __device__ __forceinline__ v8f v8f_zero() {
    v8f z = {0.f, 0.f, 0.f, 0.f, 0.f, 0.f, 0.f, 0.f};
    return z;
}

// D = A(16x4, f32) * B(4x16, f32) + C  — fp32 WMMA, wave32
static __device__ __forceinline__ v8f wmma_f32_4(v2f a, v2f b, v8f c) {
    return __builtin_amdgcn_wmma_f32_16x16x4_f32(
        /*neg_a=*/false, a, /*neg_b=*/false, b,
        /*c_mod=*/(short)0, c, /*reuse_a=*/false, /*reuse_b=*/false);
}

// ---------------------------------------------------------------------------
// TDM: DMA a contiguous block of `nelem` dwords from global into LDS.
// D# per CDNA5 ISA: 2D tensor, 1 row of nelem elements,
// data_size=4B (code 2), count=1, type=2 ("image"). Issued by one wave,
// completion via S_WAIT_TENSORCNT, then workgroup barrier.
// ---------------------------------------------------------------------------
#if defined(__has_builtin)
#if __has_builtin(__builtin_amdgcn_tensor_load_to_lds) && \
    __has_builtin(__builtin_amdgcn_s_wait_tensorcnt)
#define HAVE_TDM 1
#endif
#endif

#ifdef HAVE_TDM
typedef __attribute__((ext_vector_type(4))) unsigned int tdm_v4u;
typedef __attribute__((ext_vector_type(8))) int          tdm_v8i;
typedef __attribute__((ext_vector_type(4))) int          tdm_v4i;

static __device__ __forceinline__ void tdm_copy_to_lds(const float* gsrc,
                                                       float* lds_dst,
                                                       int nelem) {
    const unsigned long long ga = (unsigned long long)(uintptr_t)gsrc;
    // generic LDS pointer: low 32 bits are the wave-relative LDS byte offset
    const unsigned int la = (unsigned int)(uintptr_t)lds_dst;

    tdm_v4u g0;
    g0[0] = 1u;                                              // count=1, user mode
    g0[1] = la;                                              // lds_addr
    g0[2] = (unsigned int)(ga & 0xFFFFFFFFull);              // global_addr[31:0]
    g0[3] = (unsigned int)((ga >> 32) & 0x1FFFFFFull)        // global_addr[56:32]
            | (2u << 30);                                    // type = 2

    const unsigned int ne = (unsigned int)nelem;
    tdm_v8i g1;
    g1[0] = (int)(2u << 16);                 // wg_mask=0 | data_size=2 (4 bytes)
    g1[1] = (int)((ne & 0xFFFFu) << 16);     // abar_addr=0 | tensor_dim0[15:0]
    g1[2] = (int)((ne >> 16) | (1u << 16));  // tensor_dim0[31:16] | tensor_dim1=1
    g1[3] = (int)((ne & 0xFFFFu) << 16);     // tensor_dim1[31:16]=0 | tile_dim0
    g1[4] = 1;                               // tile_dim1=1 | tile_dim2=0
    g1[5] = (int)ne;                         // tensor_dim0_stride[31:0]
    g1[6] = 0;                               // stride0[47:32]=0 | stride1 lo
    g1[7] = 0;

    tdm_v4i z4 = {0, 0, 0, 0};
#if __clang_major__ >= 23
    tdm_v8i z8 = {0, 0, 0, 0, 0, 0, 0, 0};
    __builtin_amdgcn_tensor_load_to_lds(g0, g1, z4, z4, z8, 0);
#else
    __builtin_amdgcn_tensor_load_to_lds(g0, g1, z4, z4, 0);
#endif
    __builtin_amdgcn_s_wait_tensorcnt(0);
}
#endif

// Fill LDS with `nelem` dwords of weight data, then barrier.
static __device__ __forceinline__ void stage_weights(const float* __restrict__ gsrc,
                                                     float* lds, int nelem,
                                                     int tid) {
#ifdef HAVE_TDM
    if ((tid >> 5) == 0) {          // wave 0 issues the DMA (TDM ignores EXEC)
        tdm_copy_to_lds(gsrc, lds, nelem);
    }
#else
    for (int i = tid; i < nelem; i += 256) lds[i] = gsrc[i];
#endif
    __syncthreads();
}

// ---------------------------------------------------------------------------
// Pair-interleave swizzle so a B fragment is ONE aligned ds_load_b64:
//   dst[(k>>1)*128 + n*2 + (k&1)] = src[k*64 + n]
// ---------------------------------------------------------------------------
__global__ void swizzle_w_kernel(const float* __restrict__ src,
                                 float* __restrict__ dst, int K) {
    int t = blockIdx.x * blockDim.x + threadIdx.x;
    if (t >= K * 64) return;
    const int k = t >> 6;
    const int n = t & 63;
    dst[(k >> 1) * 128 + n * 2 + (k & 1)] = src[t];
}

// B fragment (4x16 f32) from swizzled LDS weights: single b64 load
static __device__ __forceinline__ v2f bfrag_ld(const float* Ws, int ka, int n) {
    return *(const v2f*)(Ws + ((ka >> 1) * 128 + n * 2));
}

// ---------------------------------------------------------------------------
// x[m][0:64] = feat[m][0:128] @ W[128][64] + b + emb[m]
// 256 threads = 8 waves; each wave computes one 16-row x 64-col tile.
// ---------------------------------------------------------------------------
__global__ void node_transform_kernel(const float* __restrict__ feat,
                                      const float* __restrict__ Wswz,
                                      const float* __restrict__ bias,
                                      const float* __restrict__ emb,
                                      float* __restrict__ xout, int M) {
    __shared__ float Ws[128 * 64];
    const int tid  = threadIdx.x;
    const int lane = tid & 31;
    const int wave = tid >> 5;
    stage_weights(Wswz, Ws, 128 * 64, tid);

    const int tileM = (blockIdx.x * 8 + wave) * 16;
    if (tileM >= M) return;                 // wave-uniform: EXEC stays full

    const int l16  = lane & 15;
    const int halb = lane >> 4;             // K sub-pair select (wave32 halves)
    int row = tileM + l16;
    if (row >= M) row = M - 1;              // clamp loads; stores masked below
    const float* arow = feat + (size_t)row * 128;

    v8f acc[4];
#pragma unroll
    for (int j = 0; j < 4; ++j) acc[j] = v8f_zero();

    for (int k0 = 0; k0 < 128; k0 += 4) {
        const int ka = k0 + 2 * halb;
        v2f afrag = *(const v2f*)(arow + ka);
#pragma unroll
        for (int j = 0; j < 4; ++j)
            acc[j] = wmma_f32_4(afrag, bfrag_ld(Ws, ka, 16 * j + l16), acc[j]);
    }

#pragma unroll
    for (int j = 0; j < 4; ++j) {
        const int n  = 16 * j + l16;
        const float bn = bias[n];
#pragma unroll
        for (int r = 0; r < 8; ++r) {
            const int m = tileM + r + 8 * halb;
            if (m < M)
                xout[(size_t)m * 64 + n] = acc[j][r] + bn + emb[(size_t)m * 64 + n];
        }
    }
}

// ---------------------------------------------------------------------------
// h = x @ Wc  for [N x 64] * [64 x 64]  (bias is added post-scatter)
// ---------------------------------------------------------------------------
__global__ void gcn_gemm_kernel(const float* __restrict__ x,
                                const float* __restrict__ Wswz,
                                float* __restrict__ h, int N) {
    __shared__ float Ws[64 * 64];
    const int tid  = threadIdx.x;
    const int lane = tid & 31;
    const int wave = tid >> 5;
    stage_weights(Wswz, Ws, 64 * 64, tid);

    const int tileM = (blockIdx.x * 8 + wave) * 16;
    if (tileM >= N) return;

    const int l16  = lane & 15;
    const int halb = lane >> 4;
    int row = tileM + l16;
    if (row >= N) row = N - 1;
    const float* arow = x + (size_t)row * 64;

    v8f acc[4];
#pragma unroll
    for (int j = 0; j < 4; ++j) acc[j] = v8f_zero();

    for (int k0 = 0; k0 < 64; k0 += 4) {
        const int ka = k0 + 2 * halb;
        v2f afrag = *(const v2f*)(arow + ka);
#pragma unroll
        for (int j = 0; j < 4; ++j)
            acc[j] = wmma_f32_4(afrag, bfrag_ld(Ws, ka, 16 * j + l16), acc[j]);
    }

#pragma unroll
    for (int j = 0; j < 4; ++j) {
        const int n = 16 * j + l16;
#pragma unroll
        for (int r = 0; r < 8; ++r) {
            const int m = tileM + r + 8 * halb;
            if (m < N) h[(size_t)m * 64 + n] = acc[j][r];
        }
    }
}

// ---------------------------------------------------------------------------
// Degrees (self-loop baked in as init=1.0), then dis = rsqrt(deg) in place
// ---------------------------------------------------------------------------
__global__ void deg_init_kernel(float* __restrict__ deg, int N) {
    int i = blockIdx.x * blockDim.x + threadIdx.x;
    if (i < N) deg[i] = 1.0f;               // self-loop contribution
}

__global__ void deg_accum_kernel(const long long* __restrict__ ui,
                                 const long long* __restrict__ pi,
                                 float* __restrict__ deg, int E) {
    int e = blockIdx.x * blockDim.x + threadIdx.x;
    if (e >= E) return;
    atomicAdd(deg + (int)pi[e], 1.0f);      // col = prod_idx (raw, no offset!)
    atomicAdd(deg + (int)ui[e], 1.0f);      // col = user_idx (reverse edge)
}

__global__ void dis_kernel(float* __restrict__ deg, int N) {
    int i = blockIdx.x * blockDim.x + threadIdx.x;
    if (i < N) deg[i] = rsqrtf(deg[i]);     // deg >= 1 always
}

// out[i][f] = b[f] + dis[i]^2 * h[i][f]     (bias + self-loop message)
__global__ void scatter_init_kernel(const float* __restrict__ h,
                                    const float* __restrict__ dis,
                                    const float* __restrict__ bias,
                                    float* __restrict__ out, int N) {
    int t = blockIdx.x * blockDim.x + threadIdx.x;
    if (t >= N * 64) return;
    const int f = t & 63;
    const int i = t >> 6;
    const float d = dis[i];
    out[t] = bias[f] + d * d * h[t];
}

// symmetric message passing: out[p] += n*h[u]; out[u] += n*h[p]
__global__ void scatter_edges_kernel(const long long* __restrict__ ui,
                                     const long long* __restrict__ pi,
                                     const float* __restrict__ h,
                                     const float* __restrict__ dis,
                                     float* __restrict__ out, int E) {
    long long t = (long long)blockIdx.x * blockDim.x + threadIdx.x;
    const int f = (int)(t & 63);
    const long long e = t >> 6;
    if (e >= E) return;
    if (f == 0) {                            // stream index arrays into GL2
        __builtin_prefetch((const char*)(ui + e) + 4096, 0, 1);
        __builtin_prefetch((const char*)(pi + e) + 4096, 0, 1);
    }
    const int u = (int)ui[e];
    const int p = (int)pi[e];               // raw product index (reference quirk)
    const float norm = dis[u] * dis[p];
    const float hu = h[(size_t)u * 64 + f];
    const float hp = h[(size_t)p * 64 + f];
    atomicAdd(out + (size_t)p * 64 + f, norm * hu);
    atomicAdd(out + (size_t)u * 64 + f, norm * hp);
}

__global__ void relu_kernel(float* __restrict__ x, int n) {
    int t = blockIdx.x * blockDim.x + threadIdx.x;
    if (t < n) x[t] = fmaxf(x[t], 0.0f);
}

// ---------------------------------------------------------------------------
// Predictor: per 16-edge tile, gather pair[16x128], WMMA vs W1[128x64],
// bias+relu, 64-dot with W2 via wave32 half-shuffle reduction, sigmoid*5.
// ---------------------------------------------------------------------------
__global__ void pred_kernel(const long long* __restrict__ ui,
                            const long long* __restrict__ pi,
                            const float* __restrict__ x2,
                            const float* __restrict__ W1swz,
                            const float* __restrict__ b1,
                            const float* __restrict__ W2,
                            const float* __restrict__ b2,
                            float* __restrict__ preds, int E) {
    __shared__ float W1s[128 * 64];
    const int tid  = threadIdx.x;
    const int lane = tid & 31;
    const int wave = tid >> 5;
    stage_weights(W1swz, W1s, 128 * 64, tid);

    const int tileE = (blockIdx.x * 8 + wave) * 16;
    if (tileE >= E) return;                 // E % 16 == 0, full tiles only

    const int l16  = lane & 15;
    const int halb = lane >> 4;
    const int e = tileE + l16;
    const int u = (int)ui[e];
    const int p = (int)pi[e];
    const float* rowu = x2 + (size_t)u * 64;
    const float* rowp = x2 + (size_t)(NUM_USERS + p) * 64;   // offset here!

    v8f acc[4];
#pragma unroll
    for (int j = 0; j < 4; ++j) acc[j] = v8f_zero();

    for (int k0 = 0; k0 < 128; k0 += 4) {
        const int ka = k0 + 2 * halb;                  // never straddles k=64
        const float* src = (ka < 64) ? rowu : rowp;
        v2f afrag = *(const v2f*)(src + (ka & 63));
#pragma unroll
        for (int j = 0; j < 4; ++j)
            acc[j] = wmma_f32_4(afrag, bfrag_ld(W1s, ka, 16 * j + l16), acc[j]);
    }

    float w2v[4], b1v[4];
#pragma unroll
    for (int j = 0; j < 4; ++j) {
        w2v[j] = W2[16 * j + l16];
        b1v[j] = b1[16 * j + l16];
    }
    const float b2v = b2[0];

#pragma unroll
    for (int r = 0; r < 8; ++r) {
        float s = 0.0f;
#pragma unroll
        for (int j = 0; j < 4; ++j) {
            float hv = fmaxf(acc[j][r] + b1v[j], 0.0f);
            s += hv * w2v[j];
        }
        // reduce across the 16 lanes of each wave32 half (xor <= 8 stays in half)
        s += __shfl_xor(s, 8, 32);
        s += __shfl_xor(s, 4, 32);
        s += __shfl_xor(s, 2, 32);
        s += __shfl_xor(s, 1, 32);
        if (l16 == 0) {
            const int m = tileE + r + 8 * halb;
            const float logit = s + b2v;
            preds[m] = 5.0f / (1.0f + expf(-logit));
        }
    }
}

// ---------------------------------------------------------------------------
extern "C" void kernel_launch(void* const* d_in, const int* in_sizes, int n_in,
                              void* d_out, int out_size, void* d_ws, size_t ws_size,
                              hipStream_t stream) {
    const long long* edge = (const long long*)d_in[0];
    const int E = in_sizes[0] / 2;
    const long long* ui = edge;
    const long long* pi = edge + E;

    const float* user_features = (const float*)d_in[1];
    const float* prod_features = (const float*)d_in[2];
    const float* user_emb      = (const float*)d_in[3];
    const float* prod_emb      = (const float*)d_in[4];
    const float* W_uf   = (const float*)d_in[5];
    const float* b_uf   = (const float*)d_in[6];
    const float* W_pf   = (const float*)d_in[7];
    const float* b_pf   = (const float*)d_in[8];
    const float* conv1W = (const float*)d_in[9];
    const float* conv1b = (const float*)d_in[10];
    const float* conv2W = (const float*)d_in[11];
    const float* conv2b = (const float*)d_in[12];
    const float* predW1 = (const float*)d_in[13];
    const float* predb1 = (const float*)d_in[14];
    const float* predW2 = (const float*)d_in[15];
    const float* predb2 = (const float*)d_in[16];
    float* preds = (float*)d_out;

    const int Nu = NUM_USERS, Np = NUM_PRODS, Nn = N_NODES;

    float* buf0 = (float*)d_ws;                        // node state  [Nn*64]
    float* buf1 = buf0 + (size_t)Nn * 64;              // gemm output [Nn*64]
    float* deg  = buf1 + (size_t)Nn * 64;              // deg -> dis  [Nn]
    float* wswz = deg + (size_t)Nn;                    // swizzled weights
    float* wuf_s = wswz;                               // 8192
    float* wpf_s = wuf_s + 8192;                       // 8192
    float* w1c_s = wpf_s + 8192;                       // 4096
    float* w2c_s = w1c_s + 4096;                       // 4096
    float* wp1_s = w2c_s + 4096;                       // 8192

    const int B = 256;
    const int elemN = Nn * 64;

    // 0) pair-interleave swizzle of all GEMM weights (tiny, one-shot)
    swizzle_w_kernel<<<(8192 + B - 1) / B, B, 0, stream>>>(W_uf,   wuf_s, 128);
    swizzle_w_kernel<<<(8192 + B - 1) / B, B, 0, stream>>>(W_pf,   wpf_s, 128);
    swizzle_w_kernel<<<(4096 + B - 1) / B, B, 0, stream>>>(conv1W, w1c_s, 64);
    swizzle_w_kernel<<<(4096 + B - 1) / B, B, 0, stream>>>(conv2W, w2c_s, 64);
    swizzle_w_kernel<<<(8192 + B - 1) / B, B, 0, stream>>>(predW1, wp1_s, 128);

    // 1) feature transforms -> buf0
    node_transform_kernel<<<(Nu + 127) / 128, B, 0, stream>>>(
        user_features, wuf_s, b_uf, user_emb, buf0, Nu);
    node_transform_kernel<<<(Np + 127) / 128, B, 0, stream>>>(
        prod_features, wpf_s, b_pf, prod_emb, buf0 + (size_t)Nu * 64, Np);

    // 2) degrees + inverse-sqrt
    deg_init_kernel<<<(Nn + B - 1) / B, B, 0, stream>>>(deg, Nn);
    deg_accum_kernel<<<(E + B - 1) / B, B, 0, stream>>>(ui, pi, deg, E);
    dis_kernel<<<(Nn + B - 1) / B, B, 0, stream>>>(deg, Nn);

    // 3) conv1: gemm -> scatter(+bias,self-loop) -> relu
    gcn_gemm_kernel<<<(Nn + 127) / 128, B, 0, stream>>>(buf0, w1c_s, buf1, Nn);
    scatter_init_kernel<<<(elemN + B - 1) / B, B, 0, stream>>>(buf1, deg, conv1b, buf0, Nn);
    scatter_edges_kernel<<<(int)(((long long)E * 64 + B - 1) / B), B, 0, stream>>>(
        ui, pi, buf1, deg, buf0, E);
    relu_kernel<<<(elemN + B - 1) / B, B, 0, stream>>>(buf0, elemN);

    // 4) conv2
    gcn_gemm_kernel<<<(Nn + 127) / 128, B, 0, stream>>>(buf0, w2c_s, buf1, Nn);
    scatter_init_kernel<<<(elemN + B - 1) / B, B, 0, stream>>>(buf1, deg, conv2b, buf0, Nn);
    scatter_edges_kernel<<<(int)(((long long)E * 64 + B - 1) / B), B, 0, stream>>>(
        ui, pi, buf1, deg, buf0, E);

    // 5) predictor
    const int tiles  = (E + 15) / 16;
    const int blocks = (tiles + 7) / 8;
    pred_kernel<<<blocks, B, 0, stream>>>(ui, pi, buf0, wp1_s, predb1,
                                          predW2, predb2, preds, E);
}